// DP_44315472560786
// MI455X (gfx1250) — compile-verified
//
#include <hip/hip_runtime.h>
#include <stdint.h>

// Problem constants (match reference)
#define B_      32
#define NTYPES_ 2
#define NATPT_  128
#define N_      (NTYPES_ * NATPT_)   // 256
#define M_      200
#define RMIN_   5.8f
#define RMAX_   6.0f
#define DU_     5.0f                 // 1/(RMAX-RMIN)

// LDS layout (bytes), every segment 16B aligned for b128 async targets.
// After the compute phase consumes all inputs into registers, bytes [0,9600)
// are reused as the Ri_d staging buffer (200 * 48 B) for dense async stores.
#define LOFF_DR    0      // image_dR slab: 600 f32 (2400 B)
#define LOFF_RX    2400   // Ri_xyz  slab: 600 f32 (2400 B)
#define LOFF_X     4800   // x:   200 f32 (800 B)
#define LOFF_INR   5600   // inr: 200 f32 (800 B)
#define LOFF_DAVG  6400   // davg row: 800 f32 (3200 B)
#define LOFF_DSTD  9600   // dstd row: 800 f32 (3200 B)
#define LOFF_MASK  12800  // mask: 200 bytes (pad to 208)
#define SMEM_BYTES 13056

__device__ __forceinline__ void async_ld_b128(uint32_t lds_byte_addr, const void* gaddr) {
  asm volatile("global_load_async_to_lds_b128 %0, %1, off"
               :: "v"(lds_byte_addr), "v"(gaddr) : "memory");
}
__device__ __forceinline__ void async_ld_b32(uint32_t lds_byte_addr, const void* gaddr) {
  asm volatile("global_load_async_to_lds_b32 %0, %1, off"
               :: "v"(lds_byte_addr), "v"(gaddr) : "memory");
}
__device__ __forceinline__ void async_st_b128(const void* gaddr, uint32_t lds_byte_addr) {
  asm volatile("global_store_async_from_lds_b128 %0, %1, off"
               :: "v"(gaddr), "v"(lds_byte_addr) : "memory");
}
__device__ __forceinline__ void wait_async0() {
  asm volatile("s_wait_asynccnt 0" ::: "memory");
}

__global__ __launch_bounds__(256)
void dp_smooth_desc_kernel(const float* __restrict__ image_dR,
                           const float* __restrict__ x_in,
                           const float* __restrict__ Ri_xyz,
                           const uint8_t* __restrict__ mask,   // numpy bool, 1 byte/elem
                           const float* __restrict__ inr_in,
                           const float* __restrict__ davg,     // [NTYPES, M*4]
                           const float* __restrict__ dstd,     // [NTYPES, M*4]
                           float* __restrict__ out)            // Ri flat ++ Ri_d flat
{
  __shared__ __align__(16) uint8_t smem[SMEM_BYTES];
  const int tid = (int)threadIdx.x;
  const int bn  = (int)blockIdx.x;              // 0 .. B*N-1
  const int n   = bn & (N_ - 1);
  const int typ = n >> 7;                       // n / NATPT_

  // Low 32 bits of a generic pointer to __shared__ == LDS byte address
  const uint32_t lds0 = (uint32_t)(uintptr_t)(void*)smem;

  const float*   dR_g  = image_dR + (size_t)bn * (M_ * 3);
  const float*   rx_g  = Ri_xyz   + (size_t)bn * (M_ * 3);
  const float*   x_g   = x_in     + (size_t)bn * M_;
  const float*   inr_g = inr_in   + (size_t)bn * M_;
  const uint8_t* mk_g  = mask     + (size_t)bn * M_;
  const float*   av_g  = davg     + (size_t)typ * (M_ * 4);
  const float*   sd_g  = dstd     + (size_t)typ * (M_ * 4);

  // ---- Stage all per-(b,n) inputs into LDS: straight-line, predicated,
  //      one async B128 per segment per thread (all segment counts <= 256) ----
  if (tid < 150) async_ld_b128(lds0 + LOFF_DR   + (uint32_t)tid * 16u, dR_g  + tid * 4);
  if (tid < 150) async_ld_b128(lds0 + LOFF_RX   + (uint32_t)tid * 16u, rx_g  + tid * 4);
  if (tid <  50) async_ld_b128(lds0 + LOFF_X    + (uint32_t)tid * 16u, x_g   + tid * 4);
  if (tid <  50) async_ld_b128(lds0 + LOFF_INR  + (uint32_t)tid * 16u, inr_g + tid * 4);
  if (tid < 200) async_ld_b128(lds0 + LOFF_DAVG + (uint32_t)tid * 16u, av_g  + tid * 4);
  if (tid < 200) async_ld_b128(lds0 + LOFF_DSTD + (uint32_t)tid * 16u, sd_g  + tid * 4);
  if (tid <  50) async_ld_b32 (lds0 + LOFF_MASK + (uint32_t)tid * 4u,  mk_g  + tid * 4);

  wait_async0();
  __syncthreads();

  // ---- Compute: one neighbor (m) per thread, all in registers ----
  const bool active = (tid < M_);
  float rd[12];
  float4 riOut;
  if (active) {
    const int m = tid;
    const float*   sdR  = (const float*)(smem + LOFF_DR);
    const float*   srx  = (const float*)(smem + LOFF_RX);
    const float*   sx   = (const float*)(smem + LOFF_X);
    const float*   sinr = (const float*)(smem + LOFF_INR);
    const float4*  sav  = (const float4*)(smem + LOFF_DAVG);
    const float4*  ssd  = (const float4*)(smem + LOFF_DSTD);
    const uint8_t* smk  = (const uint8_t*)(smem + LOFF_MASK);

    const float dx = sdR[3 * m + 0];
    const float dy = sdR[3 * m + 1];
    const float dz = sdR[3 * m + 2];
    const float r1 = srx[3 * m + 0];
    const float r2 = srx[3 * m + 1];
    const float r3 = srx[3 * m + 2];
    const float xv = sx[m];
    const float iv = sinr[m];
    const float4 av = sav[m];
    const float4 sd = ssd[m];
    const bool  mk = (smk[m] != 0);
    const float mf = mk ? 1.0f : 0.0f;

    // quintic switching function vv(x) and derivative dvv(x)
    const float uu   = (xv - RMIN_) * DU_;
    const float uu2  = uu * uu;
    const float uu3  = uu2 * uu;
    const float pc   = fmaf(-6.0f, uu2, fmaf(15.0f, uu, -10.0f));
    const float poly = fmaf(uu3, pc, 1.0f);
    const float dpoly = (3.0f * uu2 * pc + uu3 * fmaf(-12.0f, uu, 15.0f)) * DU_;
    const bool  inmid = mk && (xv >= RMIN_) && (xv < RMAX_);
    const float vv  = (mk && (xv < RMIN_)) ? 1.0f : (inmid ? poly : 0.0f);
    const float dvv = inmid ? dpoly : 0.0f;

    const float inr_m = iv * mf;           // masked 1/r
    const float inr2  = inr_m * inr_m;
    const float inr4  = inr2 * inr2;
    const float inr3  = inr4 * xv * mf;    // masked 1/r^3
    const float res   = inr_m;             // Ri[...,0] pre-scale
    const float common = dvv * inr_m;      // dvv * (1/r), masked

    // one reciprocal per dstd component instead of 16 true divisions
    const float is0 = 1.0f / sd.x;
    const float is1 = 1.0f / sd.y;
    const float is2 = 1.0f / sd.z;
    const float is3 = 1.0f / sd.w;

    // row 0: dR_k * (inr3*vv - res*common)
    const float c0 = inr3 * vv - res * common;
    rd[0] = dx * c0 * is0;
    rd[1] = dy * c0 * is0;
    rd[2] = dz * c0 * is0;
    // rows 1..3: 2*dRj*dRk*inr4*vv - delta_jk*inr2*vv - Rj*common*dRk
    const float a  = 2.0f * inr4 * vv;
    const float bt = inr2 * vv;
    const float Rj[3]  = { r1, r2, r3 };
    const float Dk[3]  = { dx, dy, dz };
    const float isv[3] = { is1, is2, is3 };
#pragma unroll
    for (int j = 0; j < 3; ++j) {
      const float cj = Rj[j] * common;
#pragma unroll
      for (int k = 0; k < 3; ++k) {
        float v = Dk[j] * Dk[k] * a - cj * Dk[k];
        if (j == k) v -= bt;
        rd[3 + j * 3 + k] = v * isv[j];    // masked factors -> 0 when !mk
      }
    }

    // Ri scaled by vv only where masked; unmasked entries untouched
    const float sv = mk ? vv : 1.0f;
    riOut = make_float4((res * sv - av.x) * is0,
                        (r1  * sv - av.y) * is1,
                        (r2  * sv - av.z) * is2,
                        (r3  * sv - av.w) * is3);
  }

  // All LDS input reads are done; reuse bytes [0,9600) as Ri_d staging.
  __syncthreads();

  if (active) {
    // Ri: dense per-lane float4, perfectly coalesced direct store
    float4* outRi = (float4*)out;                       // Ri: [B,N,M,4]
    outRi[(size_t)bn * M_ + tid] = riOut;

    // Scatter this thread's 12 Ri_d floats into LDS in final memory order
    float4* srd = (float4*)smem;                        // [200*3] float4
    srd[tid * 3 + 0] = make_float4(rd[0], rd[1], rd[2],  rd[3]);
    srd[tid * 3 + 1] = make_float4(rd[4], rd[5], rd[6],  rd[7]);
    srd[tid * 3 + 2] = make_float4(rd[8], rd[9], rd[10], rd[11]);
  }

  __syncthreads();

  // Drain Ri_d slab (9600 B = 600 x 16 B) with dense async LDS->global stores
  {
    const float* outd = out + (size_t)B_ * N_ * M_ * 4 + (size_t)bn * (M_ * 12);
    if (true) {
      // chunks tid, tid+256, tid+512 (<600)
      async_st_b128((const void*)((const uint8_t*)outd + (uint32_t)tid * 16u),
                    lds0 + (uint32_t)tid * 16u);
      if (tid + 256 < 600)
        async_st_b128((const void*)((const uint8_t*)outd + ((uint32_t)tid + 256u) * 16u),
                      lds0 + ((uint32_t)tid + 256u) * 16u);
      if (tid + 512 < 600)
        async_st_b128((const void*)((const uint8_t*)outd + ((uint32_t)tid + 512u) * 16u),
                      lds0 + ((uint32_t)tid + 512u) * 16u);
    }
  }
  wait_async0();   // (S_ENDPGM also implies wait-idle; explicit for clarity)
}

extern "C" void kernel_launch(void* const* d_in, const int* in_sizes, int n_in,
                              void* d_out, int out_size, void* d_ws, size_t ws_size,
                              hipStream_t stream) {
  (void)in_sizes; (void)n_in; (void)out_size; (void)d_ws; (void)ws_size;
  const float*   image_dR = (const float*)d_in[0];
  const float*   x        = (const float*)d_in[1];
  const float*   Ri_xyz   = (const float*)d_in[2];
  const uint8_t* mask     = (const uint8_t*)d_in[3];   // numpy bool -> 1 byte/elem
  const float*   inr      = (const float*)d_in[4];
  const float*   davg     = (const float*)d_in[5];
  const float*   dstd     = (const float*)d_in[6];
  // d_in[7] = natoms_per_type (python int); constant 128, baked in
  float* out = (float*)d_out;

  dp_smooth_desc_kernel<<<B_ * N_, 256, 0, stream>>>(
      image_dR, x, Ri_xyz, mask, inr, davg, dstd, out);
}